// ResidualStruct3Model_27625229648377
// MI455X (gfx1250) — compile-verified
//
#include <hip/hip_runtime.h>

// ---------------------------------------------------------------------------
// MI455X (gfx1250) implementation.
//  Kernel 1: streaming 7->1 projection + sliding max-pool (memory bound,
//            552 MB of x traffic -> vectorized b128 loads via LDS staging).
//  Kernel 2: fused MLP (600->128 ->res128 ->32 ->2) using
//            V_WMMA_F32_16X16X4_F32, one wave per 16-row tile, intermediates
//            in padded LDS, weights/feat served from L2 (192 MB).
// ---------------------------------------------------------------------------

typedef float v2f __attribute__((ext_vector_type(2)));
typedef float v8f __attribute__((ext_vector_type(8)));

#define WMMA4(a, b, c) \
  __builtin_amdgcn_wmma_f32_16x16x4_f32(false, (a), false, (b), (short)0, (c), false, false)

static constexpr int QS    = 1204;
static constexpr int FD    = 7;
static constexpr int ROWF  = QS * FD;   // 8428 floats per batch row (16B-aligned rows)
static constexpr int ROWF4 = ROWF / 4;  // 2107 float4 per row
static constexpr int NFEAT = 600;
static constexpr int LSTR  = 132;       // LDS row stride (pad: 132 % 64 = 4 -> no bank conflicts)

// ---------------------------------------------------------------------------
// Kernel 1: s[b,q] = dot(x[b,q,:], w_step) + b_step ; feat = sliding maxpool
// One workgroup per batch row. Stage the 33.7 KB row through LDS with float4
// loads for clean global_load_b128 streaming.
// ---------------------------------------------------------------------------
__global__ __launch_bounds__(256) void step_pool_kernel(
    const float* __restrict__ x, const float* __restrict__ w_step,
    const float* __restrict__ b_step, float* __restrict__ feat)
{
  __shared__ float sx[ROWF];
  __shared__ float ss[QS];
  const int b   = blockIdx.x;
  const int tid = threadIdx.x;

  const float4* __restrict__ xrow = reinterpret_cast<const float4*>(x + (size_t)b * ROWF);
  float4* sx4 = reinterpret_cast<float4*>(sx);
  #pragma unroll 4
  for (int i = tid; i < ROWF4; i += 256) sx4[i] = xrow[i];

  const float w0 = w_step[0], w1 = w_step[1], w2 = w_step[2], w3 = w_step[3],
              w4 = w_step[4], w5 = w_step[5], w6 = w_step[6], bs = b_step[0];
  __syncthreads();

  for (int q = tid; q < QS; q += 256) {
    const float* p = sx + q * FD;
    float a = bs;
    a = fmaf(p[0], w0, a); a = fmaf(p[1], w1, a); a = fmaf(p[2], w2, a);
    a = fmaf(p[3], w3, a); a = fmaf(p[4], w4, a); a = fmaf(p[5], w5, a);
    a = fmaf(p[6], w6, a);
    ss[q] = a;
  }
  __syncthreads();

  // feat[b, bl*5 + off] = max_{i<10} s[b, off + bl*10 + i]
  float* frow = feat + (size_t)b * NFEAT;
  for (int j = tid; j < NFEAT; j += 256) {
    const int bl = j / 5, off = j % 5;
    const float* p = ss + off + bl * 10;
    float m = p[0];
    #pragma unroll
    for (int i = 1; i < 10; ++i) m = fmaxf(m, p[i]);
    frow[j] = m;
  }
}

// ---------------------------------------------------------------------------
// WMMA fragment helpers for V_WMMA_F32_16X16X4_F32 (wave32).
// A (16x4 f32): lanes 0-15 hold row M=lane,    VGPR0=K+0, VGPR1=K+1
//               lanes 16-31 hold row M=lane-16, VGPR0=K+2, VGPR1=K+3
// B (4x16):     assumed symmetric (col N = lane&15, same K split).
// D (16x16):    lane&15 = N column, VGPR i -> M = i + 8*(lane>=16).
// ---------------------------------------------------------------------------
__device__ __forceinline__ v2f load_frag_g(const float* __restrict__ p) {
  const float2 t = *reinterpret_cast<const float2*>(p);
  v2f a; a.x = t.x; a.y = t.y; return a;
}
__device__ __forceinline__ v2f load_frag_s(const float* p) {
  const float2 t = *reinterpret_cast<const float2*>(p);
  v2f a; a.x = t.x; a.y = t.y; return a;
}

// ---------------------------------------------------------------------------
// Kernel 2: fused MLP. 2 waves per block, each wave owns 16 batch rows.
// feat (39 MB) and all weights (~457 KB) are L2-resident after kernel 1.
// ---------------------------------------------------------------------------
__global__ __launch_bounds__(64) void mlp_wmma_kernel(
    const float* __restrict__ feat,
    const float* __restrict__ wA, const float* __restrict__ bA,
    const float* __restrict__ w1, const float* __restrict__ b1,
    const float* __restrict__ w2, const float* __restrict__ b2,
    const float* __restrict__ wB, const float* __restrict__ bB,
    const float* __restrict__ wC, const float* __restrict__ bC,
    float* __restrict__ out)
{
  __shared__ float lds[2 * 2 * 16 * LSTR];  // 2 waves x (hA + hB) -> 33.8 KB
  const int lane    = threadIdx.x & 31;
  const int wave    = threadIdx.x >> 5;
  const int rowBase = blockIdx.x * 32 + wave * 16;

  float* hA = lds + wave * (2 * 16 * LSTR);  // 16 x LSTR scratch (A operands)
  float* hB = hA + 16 * LSTR;                // 16 x LSTR scratch

  const int rA    = lane & 15;           // A row in tile / D column index
  const int kOff  = (lane >> 4) << 1;    // +0 (lanes 0-15) or +2 (lanes 16-31)
  const int mHalf = (lane >> 4) << 3;    // D rows = mHalf + i

  // ---- stage A: h = relu(feat @ wA^T + bA)   (M=16, N=128, K=600) ----
  v8f acc[8] = {};
  const float* arow = feat + (size_t)(rowBase + rA) * NFEAT + kOff;
  for (int k = 0; k < NFEAT; k += 4) {
    const v2f a = load_frag_g(arow + k);
    #pragma unroll
    for (int t = 0; t < 8; ++t) {
      const int n = t * 16 + rA;
      const v2f bf = load_frag_g(wA + (size_t)n * NFEAT + k + kOff);
      acc[t] = WMMA4(a, bf, acc[t]);
    }
  }
  v8f accH[8];  // keep h in registers for the residual add
  #pragma unroll
  for (int t = 0; t < 8; ++t) {
    const int n  = t * 16 + rA;
    const float bv = bA[n];
    #pragma unroll
    for (int i = 0; i < 8; ++i) {
      const float v = fmaxf(acc[t][i] + bv, 0.0f);
      accH[t][i] = v;
      hA[(mHalf + i) * LSTR + n] = v;     // spill D-layout -> row-major for next A
    }
  }
  __syncthreads();

  // ---- fc1: r1 = relu(h @ w1^T + b1)   (K=128) ----
  #pragma unroll
  for (int t = 0; t < 8; ++t) acc[t] = v8f{};
  for (int k = 0; k < 128; k += 4) {
    const v2f a = load_frag_s(hA + rA * LSTR + k + kOff);
    #pragma unroll
    for (int t = 0; t < 8; ++t) {
      const int n = t * 16 + rA;
      const v2f bf = load_frag_g(w1 + n * 128 + k + kOff);
      acc[t] = WMMA4(a, bf, acc[t]);
    }
  }
  #pragma unroll
  for (int t = 0; t < 8; ++t) {
    const int n  = t * 16 + rA;
    const float bv = b1[n];
    #pragma unroll
    for (int i = 0; i < 8; ++i)
      hB[(mHalf + i) * LSTR + n] = fmaxf(acc[t][i] + bv, 0.0f);
  }
  __syncthreads();

  // ---- fc2 + residual: h = relu(h + r1 @ w2^T + b2) ----
  #pragma unroll
  for (int t = 0; t < 8; ++t) acc[t] = v8f{};
  for (int k = 0; k < 128; k += 4) {
    const v2f a = load_frag_s(hB + rA * LSTR + k + kOff);
    #pragma unroll
    for (int t = 0; t < 8; ++t) {
      const int n = t * 16 + rA;
      const v2f bf = load_frag_g(w2 + n * 128 + k + kOff);
      acc[t] = WMMA4(a, bf, acc[t]);
    }
  }
  #pragma unroll
  for (int t = 0; t < 8; ++t) {
    const int n  = t * 16 + rA;
    const float bv = b2[n];
    #pragma unroll
    for (int i = 0; i < 8; ++i)
      hA[(mHalf + i) * LSTR + n] = fmaxf(accH[t][i] + acc[t][i] + bv, 0.0f);
  }
  __syncthreads();

  // ---- stage B: h3 = relu(h @ wB^T + bB)   (N=32, K=128) ----
  v8f acc2[2] = {};
  for (int k = 0; k < 128; k += 4) {
    const v2f a = load_frag_s(hA + rA * LSTR + k + kOff);
    #pragma unroll
    for (int t = 0; t < 2; ++t) {
      const int n = t * 16 + rA;
      const v2f bf = load_frag_g(wB + n * 128 + k + kOff);
      acc2[t] = WMMA4(a, bf, acc2[t]);
    }
  }
  #pragma unroll
  for (int t = 0; t < 2; ++t) {
    const int n  = t * 16 + rA;
    const float bv = bB[n];
    #pragma unroll
    for (int i = 0; i < 8; ++i)
      hB[(mHalf + i) * 36 + n] = fmaxf(acc2[t][i] + bv, 0.0f);  // 16x32, stride 36
  }
  __syncthreads();

  // ---- stage C: logits (2 cols) — 32 lanes = 16 rows x 2 cols ----
  {
    const int row = lane & 15;
    const int col = lane >> 4;
    const float* hp = hB + row * 36;
    const float* wc = wC + col * 32;
    float s = bC[col];
    #pragma unroll
    for (int k = 0; k < 32; ++k) s = fmaf(hp[k], wc[k], s);
    out[(size_t)(rowBase + row) * 2 + col] = s;
  }
}

// ---------------------------------------------------------------------------
extern "C" void kernel_launch(void* const* d_in, const int* in_sizes, int n_in,
                              void* d_out, int out_size, void* d_ws, size_t ws_size,
                              hipStream_t stream) {
  const float* x      = (const float*)d_in[0];
  const float* w_step = (const float*)d_in[1];
  const float* b_step = (const float*)d_in[2];
  const float* wA = (const float*)d_in[3];
  const float* bA = (const float*)d_in[4];
  const float* w1 = (const float*)d_in[5];
  const float* b1 = (const float*)d_in[6];
  const float* w2 = (const float*)d_in[7];
  const float* b2 = (const float*)d_in[8];
  const float* wB = (const float*)d_in[9];
  const float* bB = (const float*)d_in[10];
  const float* wC = (const float*)d_in[11];
  const float* bC = (const float*)d_in[12];
  float* out  = (float*)d_out;
  float* feat = (float*)d_ws;                    // B * 600 floats (39.3 MB)

  const int B = in_sizes[0] / ROWF;              // 16384

  step_pool_kernel<<<B, 256, 0, stream>>>(x, w_step, b_step, feat);
  mlp_wmma_kernel<<<B / 32, 64, 0, stream>>>(feat, wA, bA, w1, b1, w2, b2,
                                             wB, bB, wC, bC, out);
}